// RoutingCapsule_74096775791307
// MI455X (gfx1250) — compile-verified
//
#include <hip/hip_runtime.h>

// ---------------------------------------------------------------------------
// RoutingCapsule on gfx1250 (MI455X).
// Math collapses to: s[b,kc,o] = sum_{n,i} x[b,n,i]*W[n,kc,i,o]; out = squash(s).
// GEMM: M=64 (batch), N=1024 (kc*32+o), K=32768 (n*16+i). f32 WMMA 16x16x4.
// ---------------------------------------------------------------------------

typedef __attribute__((ext_vector_type(2))) float v2f;
typedef __attribute__((ext_vector_type(8))) float v8f;

#define BATCH   64
#define NCAPS   2048
#define IDIM    16
#define KCLS    32
#define ODIM    32
#define COLS    (KCLS * ODIM)        // 1024 output columns
#define KRED    (NCAPS * IDIM)       // 32768 reduction depth
#define NCHUNKS 64                   // reduction split -> 64*64 = 4096 waves
#define NPER    (NCAPS / NCHUNKS)    // 32 capsules per chunk
#define WPB     8                    // waves per 256-thread block

// W[n,kc,i,o] flat = n*16384 + kc*512 + i*32 + o
#define W_NSTRIDE 16384
#define W_KSTRIDE 512
#define W_ISTRIDE 32

__global__ void caps_zero(float* __restrict__ s) {
  s[blockIdx.x * blockDim.x + threadIdx.x] = 0.0f;
}

__global__ __launch_bounds__(256) void caps_gemm(const float* __restrict__ x,
                                                 const float* __restrict__ W,
                                                 float* __restrict__ s) {
  const int lane = threadIdx.x & 31;
  const int wid  = blockIdx.x * WPB + (threadIdx.x >> 5);   // 0..4095
  const int ct   = wid & 63;          // column tile: 16 cols of (kc,o)
  const int g    = wid >> 6;          // reduction chunk: capsules [g*32, g*32+32)
  const int kc   = ct >> 1;
  const int o0   = (ct & 1) << 4;
  const int col  = lane & 15;
  const int kkb  = (lane >> 4) << 1;  // K sub-offset per ISA A/B fragment layout

  // A fragment base (x is row-major (64, 32768)); lane m = mt*16 + col.
  const float* xb = x + (size_t)col * KRED + (size_t)g * (NPER * IDIM) + kkb;
  // B fragment base: row kr=(n,i), col = kc*32 + o0 + col.
  const float* wb = W + (size_t)g * NPER * W_NSTRIDE + (size_t)kc * W_KSTRIDE
                      + (size_t)kkb * W_ISTRIDE + o0 + col;

  v8f acc0 = {}; v8f acc1 = {}; v8f acc2 = {}; v8f acc3 = {};

  for (int nn = 0; nn < NPER; ++nn) {
    const float* wn = wb + (size_t)nn * W_NSTRIDE;
    const float* xn = xb + nn * IDIM;
    // Stream W ahead (global_prefetch_b8); clamp at chunk tail.
    const float* pf = wn + ((nn < NPER - 2) ? 2 * W_NSTRIDE : 0);
    __builtin_prefetch(pf, 0, 0);
#pragma unroll
    for (int ss = 0; ss < 4; ++ss) {           // i0 = 4*ss, K step of 4
      v2f bfr;
      bfr.x = wn[ss * 4 * W_ISTRIDE];           // K = kr0+kkb
      bfr.y = wn[ss * 4 * W_ISTRIDE + W_ISTRIDE]; // K = kr0+kkb+1
      v2f a0 = *(const v2f*)(xn + ss * 4);                 // m =  0..15
      v2f a1 = *(const v2f*)(xn + ss * 4 + 16 * KRED);     // m = 16..31
      v2f a2 = *(const v2f*)(xn + ss * 4 + 32 * KRED);     // m = 32..47
      v2f a3 = *(const v2f*)(xn + ss * 4 + 48 * KRED);     // m = 48..63
      acc0 = __builtin_amdgcn_wmma_f32_16x16x4_f32(false, a0, false, bfr, (short)0, acc0, false, false);
      acc1 = __builtin_amdgcn_wmma_f32_16x16x4_f32(false, a1, false, bfr, (short)0, acc1, false, false);
      acc2 = __builtin_amdgcn_wmma_f32_16x16x4_f32(false, a2, false, bfr, (short)0, acc2, false, false);
      acc3 = __builtin_amdgcn_wmma_f32_16x16x4_f32(false, a3, false, bfr, (short)0, acc3, false, false);
    }
  }

  // C/D layout: VGPR r, lanes 0-15 -> M=r, lanes 16-31 -> M=r+8 (ISA 7.12.2).
  float* srow = s + (size_t)(ct * 16 + col);
  const int mhi = (lane >> 4) << 3;
#pragma unroll
  for (int r = 0; r < 8; ++r) {
    atomicAdd(srow + (size_t)( 0 + r + mhi) * COLS, acc0[r]);
    atomicAdd(srow + (size_t)(16 + r + mhi) * COLS, acc1[r]);
    atomicAdd(srow + (size_t)(32 + r + mhi) * COLS, acc2[r]);
    atomicAdd(srow + (size_t)(48 + r + mhi) * COLS, acc3[r]);
  }
}

// One lane per (b, kc, o); 32-lane shuffle reduction over o for the squash norm.
__global__ void caps_squash(const float* __restrict__ s, float* __restrict__ out) {
  const int t = blockIdx.x * blockDim.x + threadIdx.x;   // 0..65535
  float v = s[t];
  float q = v * v;
#pragma unroll
  for (int off = 16; off > 0; off >>= 1)
    q += __shfl_xor(q, off, 32);
  const float scale = (q / (1.0f + q)) * rsqrtf(q);
  out[t] = scale * v;
}

extern "C" void kernel_launch(void* const* d_in, const int* in_sizes, int n_in,
                              void* d_out, int out_size, void* d_ws, size_t ws_size,
                              hipStream_t stream) {
  (void)in_sizes; (void)n_in; (void)out_size; (void)ws_size;
  const float* x = (const float*)d_in[0];   // (64, 2048, 16) f32
  const float* W = (const float*)d_in[1];   // (2048, 32, 16, 32) f32
  float* out = (float*)d_out;               // (64, 32, 32) f32
  float* s   = (float*)d_ws;                // 64*1024 f32 accumulator (256 KB)

  caps_zero  <<<BATCH * COLS / 256, 256, 0, stream>>>(s);
  caps_gemm  <<<(64 * NCHUNKS) / WPB, 256, 0, stream>>>(x, W, s);
  caps_squash<<<BATCH * COLS / 256, 256, 0, stream>>>(s, out);
}